// PositionEncoder_17377437679637
// MI455X (gfx1250) — compile-verified
//
#include <hip/hip_runtime.h>
#include <hip/hip_bf16.h>

typedef float v2f __attribute__((ext_vector_type(2)));
typedef float v8f __attribute__((ext_vector_type(8)));

#define Bsz 8
#define Nn  128
#define Ee  256
#define Dd  64

__device__ __forceinline__ float envelope6(float x) {
    // P=6: EA=-28, EB=48, EC=-21 ; env = (1/x + EA x^5 + EB x^6 + EC x^7) * (x<1)
    float x2 = x * x;
    float x4 = x2 * x2;
    float x5 = x4 * x;
    float x6 = x5 * x;
    float x7 = x6 * x;
    float e = __frcp_rn(x) + (-28.0f) * x5 + 48.0f * x6 + (-21.0f) * x7;
    return (x < 1.0f) ? e : 0.0f;
}

// Branchless weight-operand build: load ALL candidates (tiny, L0-resident
// arrays) in every lane, select with v_cndmask. No pointer-ternary => no
// exec-mask divergence in the streaming loop.
__device__ __forceinline__ v2f make_bw(const float* __restrict__ W,
                                       const float* __restrict__ bias,
                                       int d, bool hi) {
    float w0 = W[d * 3 + 0];
    float w1 = W[d * 3 + 1];
    float w2 = W[d * 3 + 2];
    float bb = bias[d];
    v2f r;
    r.x = hi ? w2 : w0;
    r.y = hi ? bb : w1;
    return r;
}

// ---------------------------------------------------------------------------
// Kernel A: per-edge geometry (vec, |vec|) -> workspace, plus bdl_ebd output.
// One thread per (edge, channel); tiny kernel (131072 threads).
// ---------------------------------------------------------------------------
__global__ __launch_bounds__(256) void edge_kernel(
    const float* __restrict__ pos, const float* __restrict__ freq,
    const float* __restrict__ W_bdl, const float* __restrict__ b_bdl,
    const int* __restrict__ src, const int* __restrict__ dst,
    float4* __restrict__ ws_edges, float* __restrict__ out_bdl)
{
    int gid = blockIdx.x * 256 + threadIdx.x;     // < B*E*D = 131072
    int e = gid >> 6;                             // flat edge index b*E + e
    int d = gid & 63;
    int b = e >> 8;
    int s = src[e];
    int t = dst[e];
    const float* ps = pos + ((size_t)b * Nn + s) * 3;
    const float* pt = pos + ((size_t)b * Nn + t) * 3;
    float vx = pt[0] - ps[0];
    float vy = pt[1] - ps[1];
    float vz = pt[2] - ps[2];
    float bdl = sqrtf(vx * vx + vy * vy + vz * vz);
    if (d == 0) ws_edges[e] = make_float4(vx, vy, vz, bdl);

    float x = bdl * 0.2f;                         // /CUTOFF
    float env = envelope6(x);
    float g0 = env * __sinf(freq[0] * x);
    float g1 = env * __sinf(freq[1] * x);
    float g2 = env * __sinf(freq[2] * x);
    float r = fmaf(W_bdl[d * 3 + 0], g0,
              fmaf(W_bdl[d * 3 + 1], g1,
              fmaf(W_bdl[d * 3 + 2], g2, b_bdl[d])));
    out_bdl[gid] = r;
}

// ---------------------------------------------------------------------------
// Kernel B: dis RBF + two 3->64 linears via V_WMMA_F32_16X16X4_F32,
// fused residual add into outputs 0 and 1. One wave = 16 spatial x 64 chan.
// ---------------------------------------------------------------------------
__global__ __launch_bounds__(256) void dis_kernel(
    const float* __restrict__ pos, const float* __restrict__ freq,
    const float* __restrict__ W_da, const float* __restrict__ b_da,
    const float* __restrict__ W_dv, const float* __restrict__ b_dv,
    const float* __restrict__ edge_att, const float* __restrict__ edge_val,
    float* __restrict__ out_att, float* __restrict__ out_val)
{
    const int lane = threadIdx.x & 31;
    const int wave = threadIdx.x >> 5;
    const bool hi  = lane >= 16;
    const int  n16 = lane & 15;

    const int s_base = (blockIdx.x * 8 + wave) * 16;   // < B*N*N = 131072
    const int s = s_base + n16;                        // this lane's spatial row
    int b   = s >> 14;           // / (N*N)
    int rem = s & 16383;
    int i   = rem >> 7;
    int j   = rem & 127;

    const float* pi = pos + ((size_t)b * Nn + i) * 3;
    const float* pj = pos + ((size_t)b * Nn + j) * 3;
    float dx = pi[0] - pj[0];
    float dy = pi[1] - pj[1];
    float dz = pi[2] - pj[2];
    float dist = sqrtf(dx * dx + dy * dy + dz * dz) + 1e-6f;
    float x = dist * 0.2f;
    float env = envelope6(x);

    // freq is uniform & tiny: load all three, select by value.
    float fq0 = freq[0], fq1 = freq[1], fq2 = freq[2];

    // A-matrix 16x4 f32: lanes0-15 -> {K0,K1}, lanes16-31 -> {K2,K3=1 (bias feat)}
    float sA = env * __sinf((hi ? fq2 : fq0) * x);
    float sB = env * __sinf(fq1 * x);
    v2f a;
    a.x = sA;
    a.y = hi ? 1.0f : sB;

    // Build all 8 weight operands branchlessly, loop fully unrolled.
    #pragma unroll
    for (int t = 0; t < 4; ++t) {
        const int d = t * 16 + n16;                 // output channel (column N)
        v2f bw_att = make_bw(W_da, b_da, d, hi);
        v2f bw_val = make_bw(W_dv, b_dv, d, hi);

        // C/D 16x16 f32: VGPR r -> row M = r (+8 for hi half), col = lane&15
        size_t base = (size_t)(s_base + (hi ? 8 : 0)) * Dd + d;

        v8f c;
        #pragma unroll
        for (int r = 0; r < 8; ++r)
            c[r] = __builtin_nontemporal_load(&edge_att[base + (size_t)r * Dd]);
        v8f res = __builtin_amdgcn_wmma_f32_16x16x4_f32(
            false, a, false, bw_att, (short)0, c, false, false);
        #pragma unroll
        for (int r = 0; r < 8; ++r)
            __builtin_nontemporal_store(res[r], &out_att[base + (size_t)r * Dd]);

        #pragma unroll
        for (int r = 0; r < 8; ++r)
            c[r] = __builtin_nontemporal_load(&edge_val[base + (size_t)r * Dd]);
        res = __builtin_amdgcn_wmma_f32_16x16x4_f32(
            false, a, false, bw_val, (short)0, c, false, false);
        #pragma unroll
        for (int r = 0; r < 8; ++r)
            __builtin_nontemporal_store(res[r], &out_val[base + (size_t)r * Dd]);
    }
}

// ---------------------------------------------------------------------------
// Kernel C: edge-pair angle SBF + two 3->64 linears via WMMA, fused residual
// add into outputs 2 and 3.  cos(arccos(clip(y))) == clip(y), so no arccos.
// ---------------------------------------------------------------------------
__global__ __launch_bounds__(256) void cos_kernel(
    const float4* __restrict__ ws_edges,
    const int* __restrict__ src, const int* __restrict__ dst,
    const float* __restrict__ W_ca, const float* __restrict__ b_ca,
    const float* __restrict__ W_cv, const float* __restrict__ b_cv,
    const float* __restrict__ e2e_att, const float* __restrict__ e2e_val,
    float* __restrict__ out_att, float* __restrict__ out_val)
{
    const int lane = threadIdx.x & 31;
    const int wave = threadIdx.x >> 5;
    const bool hi  = lane >= 16;
    const int  n16 = lane & 15;

    const int s_base = (blockIdx.x * 8 + wave) * 16;   // < B*E*E = 524288
    const int s = s_base + n16;
    int b   = s >> 16;           // / (E*E)
    int rem = s & 65535;
    int i   = rem >> 8;
    int j   = rem & 255;
    int ei  = b * Ee + i;
    int ej  = b * Ee + j;

    float4 vi = ws_edges[ei];
    float4 vj = ws_edges[ej];
    float dot = vi.x * vj.x + vi.y * vj.y + vi.z * vj.z;
    float cosv = dot * __frcp_rn(vj.w + 1e-6f) * __frcp_rn(vi.w + 1e-6f);

    int si = src[ei], sj = src[ej];
    int di = dst[ei], dj = dst[ej];
    int m = 0;
    if (si == sj && di != dj) m = 1;
    if (di == dj && si != sj) m = 1;
    if (si == dj) m = -1;
    if (di == sj) m = -1;
    float mf = (float)m;
    float am = fabsf(mf);                           // |m| in {0,1}, masks output
    float c  = fminf(fmaxf(cosv * mf, -1.0f), 1.0f);

    const float C0 = 0.28209479177387814f;
    const float C1 = 0.4886025119029199f;
    const float C2 = 0.31539156525252005f;
    // Features pre-masked by |m| (bias feature too -> whole linear masked)
    float f0 = C0 * am;
    float f1 = C1 * c * am;
    float f2 = C2 * (3.0f * c * c - 1.0f) * am;

    v2f a;
    a.x = hi ? f2 : f0;
    a.y = hi ? am : f1;

    #pragma unroll
    for (int t = 0; t < 4; ++t) {
        const int d = t * 16 + n16;
        v2f bw_att = make_bw(W_ca, b_ca, d, hi);
        v2f bw_val = make_bw(W_cv, b_cv, d, hi);

        size_t base = (size_t)(s_base + (hi ? 8 : 0)) * Dd + d;

        v8f cc;
        #pragma unroll
        for (int r = 0; r < 8; ++r)
            cc[r] = __builtin_nontemporal_load(&e2e_att[base + (size_t)r * Dd]);
        v8f res = __builtin_amdgcn_wmma_f32_16x16x4_f32(
            false, a, false, bw_att, (short)0, cc, false, false);
        #pragma unroll
        for (int r = 0; r < 8; ++r)
            __builtin_nontemporal_store(res[r], &out_att[base + (size_t)r * Dd]);

        #pragma unroll
        for (int r = 0; r < 8; ++r)
            cc[r] = __builtin_nontemporal_load(&e2e_val[base + (size_t)r * Dd]);
        res = __builtin_amdgcn_wmma_f32_16x16x4_f32(
            false, a, false, bw_val, (short)0, cc, false, false);
        #pragma unroll
        for (int r = 0; r < 8; ++r)
            __builtin_nontemporal_store(res[r], &out_val[base + (size_t)r * Dd]);
    }
}

extern "C" void kernel_launch(void* const* d_in, const int* in_sizes, int n_in,
                              void* d_out, int out_size, void* d_ws, size_t ws_size,
                              hipStream_t stream) {
    const float* pos   = (const float*)d_in[0];
    const float* freq  = (const float*)d_in[1];
    const float* W_da  = (const float*)d_in[2];
    const float* b_da  = (const float*)d_in[3];
    const float* W_dv  = (const float*)d_in[4];
    const float* b_dv  = (const float*)d_in[5];
    const float* W_ca  = (const float*)d_in[6];
    const float* b_ca  = (const float*)d_in[7];
    const float* W_cv  = (const float*)d_in[8];
    const float* b_cv  = (const float*)d_in[9];
    const float* W_bdl = (const float*)d_in[10];
    const float* b_bdl = (const float*)d_in[11];
    const float* eatt  = (const float*)d_in[12];
    const float* evalu = (const float*)d_in[13];
    const float* e2att = (const float*)d_in[14];
    const float* e2val = (const float*)d_in[15];
    const int*   src   = (const int*)d_in[16];
    const int*   dst   = (const int*)d_in[17];

    float* out = (float*)d_out;
    float* out0 = out;                 // (B,N,N,D) 8388608
    float* out1 = out + 8388608;       // (B,N,N,D)
    float* out2 = out + 16777216;      // (B,E,E,D) 33554432
    float* out3 = out + 50331648;      // (B,E,E,D)
    float* out4 = out + 83886080;      // (B,E,D)   131072

    float4* ws_edges = (float4*)d_ws;  // B*E float4 = 32 KB

    // 1) per-edge geometry + bdl_ebd (also fills workspace for kernel C)
    edge_kernel<<<(Bsz * Ee * Dd) / 256, 256, 0, stream>>>(
        pos, freq, W_bdl, b_bdl, src, dst, ws_edges, out4);

    // 2) dis path: B*N*N spatial / (8 waves * 16 rows) = 1024 blocks
    dis_kernel<<<(Bsz * Nn * Nn) / 128, 256, 0, stream>>>(
        pos, freq, W_da, b_da, W_dv, b_dv, eatt, evalu, out0, out1);

    // 3) cos path: B*E*E spatial / 128 = 4096 blocks
    cos_kernel<<<(Bsz * Ee * Ee) / 128, 256, 0, stream>>>(
        ws_edges, src, dst, W_ca, b_ca, W_cv, b_cv, e2att, e2val, out2, out3);
}